// FastLlamaQKVSupply_80290118632065
// MI455X (gfx1250) — compile-verified
//
#include <hip/hip_runtime.h>

// ---------------------------------------------------------------------------
// FastLlamaQKVSupply for MI455X (gfx1250, wave32).
// Memory-bound (~1.0 GB traffic, ~44us @ 23.3 TB/s). No GEMM -> no WMMA;
// we use the CDNA5 async global<->LDS B128 DMA path for the bulk cache-copy
// rows (data never touches VGPRs), wave32 shuffles for rotate_half / row-max,
// and non-temporal hints on all stream-once traffic so the 32x/8x-reused
// cos/sin tables (17 MB) stay resident in the 192 MB L2.
// ---------------------------------------------------------------------------

namespace {
constexpr int   kB            = 4;
constexpr int   kS            = 4096;
constexpr int   kH            = 32;
constexpr int   kHKV          = 8;
constexpr int   kD            = 128;      // 32 lanes * float4
constexpr int   kMaxLen       = 8192;
constexpr float kQMax         = 127.0f;
constexpr int   kWavesPerBlk  = 8;        // 256 threads
}

// Native vector type (HIP's float4 is a class; builtins need a real vector).
typedef float vf4 __attribute__((ext_vector_type(4)));

// Builtin parameter types (from hipcc diagnostic): pointer to int4-vector in
// AS1 (global) and AS3 (LDS).
typedef int int4v __attribute__((ext_vector_type(4)));
typedef __attribute__((address_space(1))) int4v* as1_v4i_ptr;
typedef __attribute__((address_space(3))) int4v* as3_v4i_ptr;

#if defined(__AMDGCN__) && \
    __has_builtin(__builtin_amdgcn_global_load_async_to_lds_b128) && \
    __has_builtin(__builtin_amdgcn_global_store_async_from_lds_b128)
#define HAVE_ASYNC_LDS 1
#else
#define HAVE_ASYNC_LDS 0
#endif

static __device__ __forceinline__ void wait_asynccnt0() {
#if defined(__AMDGCN__)
#if __has_builtin(__builtin_amdgcn_s_wait_asynccnt)
  __builtin_amdgcn_s_wait_asynccnt(0);
#else
  asm volatile("s_wait_asynccnt 0" ::: "memory");
#endif
#endif
}

// Stream-once accesses: non-temporal so they don't evict reused L2 lines.
static __device__ __forceinline__ vf4 nt_load(const float* p) {
  return __builtin_nontemporal_load((const vf4*)p);
}
static __device__ __forceinline__ void nt_store(float* p, vf4 v) {
  __builtin_nontemporal_store(v, (vf4*)p);
}
// Reused accesses (cos/sin): regular temporal.
static __device__ __forceinline__ vf4 rt_load(const float* p) {
  return *(const vf4*)p;
}

// RoPE on 4 consecutive elements (indices lane*4..lane*4+3 of a 128-row).
// rotate_half partner for index i is i^64, i.e. lane^16 under float4 packing.
static __device__ __forceinline__ vf4 rope4(vf4 x, vf4 c, vf4 sn, int lane) {
  const float sgn = (lane < 16) ? -1.0f : 1.0f;
  vf4 p;
  p.x = __shfl_xor(x.x, 16);
  p.y = __shfl_xor(x.y, 16);
  p.z = __shfl_xor(x.z, 16);
  p.w = __shfl_xor(x.w, 16);
  vf4 o;
  o.x = x.x * c.x + sgn * p.x * sn.x;
  o.y = x.y * c.y + sgn * p.y * sn.y;
  o.z = x.z * c.z + sgn * p.z * sn.z;
  o.w = x.w * c.w + sgn * p.w * sn.w;
  return o;
}

// --------------------------- tiny setup kernels -----------------------------

__global__ void zero_inv_kernel(int* __restrict__ inv) {
  const int i = blockIdx.x * blockDim.x + threadIdx.x;
  if (i < kMaxLen) inv[i] = 0;
}

__global__ void fill_inv_kernel(const int* __restrict__ cpos,
                                int* __restrict__ inv) {
  const int s = blockIdx.x * blockDim.x + threadIdx.x;
  if (s < kS) {
    const int p = cpos[s];
    if (p >= 0 && p < kMaxLen) inv[p] = s + 1;
  }
}

// ------------------------ q: RoPE + act_quant -------------------------------
// One wave per (b,h,s) row: 512 B contiguous in, 512 B contiguous out.

__global__ __launch_bounds__(256) void rope_q_quant_kernel(
    const float* __restrict__ q, const float* __restrict__ cosb,
    const float* __restrict__ sinb, float* __restrict__ outq) {
  const int lane = threadIdx.x & 31;
  const int wave = threadIdx.x >> 5;
  const size_t r = (size_t)blockIdx.x * kWavesPerBlk + wave;  // (b,h,s) row id
  const int s = (int)(r & (kS - 1));
  const int h = (int)((r >> 12) & (kH - 1));
  const int b = (int)(r >> 17);

  const size_t qbase  = ((((size_t)b * kS + s) * kH + h) << 7) + lane * 4;
  const size_t csbase = (((size_t)b * kS + s) << 7) + lane * 4;

  const vf4 x  = nt_load(q + qbase);       // stream-once
  const vf4 c  = rt_load(cosb + csbase);   // reused 32x across heads
  const vf4 sn = rt_load(sinb + csbase);

  const vf4 o = rope4(x, c, sn, lane);

  // row abs-max via wave32 xor-shuffle reduction
  float m = fmaxf(fmaxf(fabsf(o.x), fabsf(o.y)), fmaxf(fabsf(o.z), fabsf(o.w)));
  #pragma unroll
  for (int off = 16; off > 0; off >>= 1) m = fmaxf(m, __shfl_xor(m, off));

  const float scale = fmaxf(m * (1.0f / kQMax), 1e-8f);

  vf4 y;
  y.x = fminf(kQMax, fmaxf(-kQMax, rintf(o.x / scale))) * scale;
  y.y = fminf(kQMax, fmaxf(-kQMax, rintf(o.y / scale))) * scale;
  y.z = fminf(kQMax, fmaxf(-kQMax, rintf(o.z / scale))) * scale;
  y.w = fminf(kQMax, fmaxf(-kQMax, rintf(o.w / scale))) * scale;

  nt_store(outq + (r << 7) + lane * 4, y);  // (b,h,s,d) contiguous, stream-once
}

// ---------------- k/v cache: scatter fresh rows or copy past ----------------
// One wave per (b,h_kv,pos) cache row. inv[pos]==0 -> copy past row
// (async LDS DMA path); else RoPE(k)/copy(v) from source row inv[pos]-1.

__global__ __launch_bounds__(256) void kv_cache_kernel(
    const float* __restrict__ k, const float* __restrict__ v,
    const float* __restrict__ cosb, const float* __restrict__ sinb,
    const float* __restrict__ pk, const float* __restrict__ pv,
    const int* __restrict__ inv, float* __restrict__ outk,
    float* __restrict__ outv) {
  const int lane = threadIdx.x & 31;
  const int wave = threadIdx.x >> 5;
  const size_t r = (size_t)blockIdx.x * kWavesPerBlk + wave;  // (b,h,pos) row
  const int pos = (int)(r & (kMaxLen - 1));
  const int h   = (int)((r >> 13) & (kHKV - 1));
  const int b   = (int)(r >> 16);

  const size_t obase = (r << 7) + lane * 4;   // cache row, contiguous
  const int sv = inv[pos];                    // wave-uniform, L2-resident

  if (sv == 0) {
    // ---- copy past_k/past_v row -> cache row (not overwritten this call) ---
#if HAVE_ASYNC_LDS
    __shared__ __align__(16) char stage[kWavesPerBlk * 1024];  // 1 KB / wave
    const unsigned lbase = (unsigned)(size_t)stage + wave * 1024 + lane * 16;
    const unsigned lk = lbase;
    const unsigned lv = lbase + 512;
    __builtin_amdgcn_global_load_async_to_lds_b128(
        (as1_v4i_ptr)(size_t)(pk + obase), (as3_v4i_ptr)(size_t)lk, 0, 0);
    __builtin_amdgcn_global_load_async_to_lds_b128(
        (as1_v4i_ptr)(size_t)(pv + obase), (as3_v4i_ptr)(size_t)lv, 0, 0);
    wait_asynccnt0();  // loads complete before stores read LDS
    __builtin_amdgcn_global_store_async_from_lds_b128(
        (as1_v4i_ptr)(size_t)(outk + obase), (as3_v4i_ptr)(size_t)lk, 0, 0);
    __builtin_amdgcn_global_store_async_from_lds_b128(
        (as1_v4i_ptr)(size_t)(outv + obase), (as3_v4i_ptr)(size_t)lv, 0, 0);
    // S_ENDPGM does an implicit wait-idle; no LDS reuse, so no further wait.
#else
    nt_store(outk + obase, nt_load(pk + obase));
    nt_store(outv + obase, nt_load(pv + obase));
#endif
  } else {
    const int s = sv - 1;
    const size_t ibase  = ((((size_t)b * kS + s) * kHKV + h) << 7) + lane * 4;
    const size_t csbase = (((size_t)b * kS + s) << 7) + lane * 4;
    const vf4 xk = nt_load(k + ibase);       // stream-once
    const vf4 c  = rt_load(cosb + csbase);   // reused 8x across kv-heads
    const vf4 sn = rt_load(sinb + csbase);
    nt_store(outk + obase, rope4(xk, c, sn, lane));
    nt_store(outv + obase, nt_load(v + ibase));  // v: no RoPE, stream-once
  }
}

// ---------------------------------------------------------------------------

extern "C" void kernel_launch(void* const* d_in, const int* in_sizes, int n_in,
                              void* d_out, int out_size, void* d_ws,
                              size_t ws_size, hipStream_t stream) {
  (void)in_sizes; (void)n_in; (void)out_size; (void)ws_size;
  const float* q    = (const float*)d_in[0];
  const float* k    = (const float*)d_in[1];
  const float* v    = (const float*)d_in[2];
  const float* cosb = (const float*)d_in[3];
  const float* sinb = (const float*)d_in[4];
  const float* pk   = (const float*)d_in[5];
  const float* pv   = (const float*)d_in[6];
  const int*   cpos = (const int*)d_in[7];

  float* outq = (float*)d_out;
  float* outk = outq + (size_t)kB * kH * kS * kD;
  float* outv = outk + (size_t)kB * kHKV * kMaxLen * kD;
  int*   inv  = (int*)d_ws;  // kMaxLen ints (32 KB)

  zero_inv_kernel<<<(kMaxLen + 255) / 256, 256, 0, stream>>>(inv);
  fill_inv_kernel<<<(kS + 255) / 256, 256, 0, stream>>>(cpos, inv);

  const int qBlocks  = (kB * kH * kS) / kWavesPerBlk;        // 65536
  const int kvBlocks = (kB * kHKV * kMaxLen) / kWavesPerBlk; // 32768
  rope_q_quant_kernel<<<qBlocks, 256, 0, stream>>>(q, cosb, sinb, outq);
  kv_cache_kernel<<<kvBlocks, 256, 0, stream>>>(k, v, cosb, sinb, pk, pv, inv,
                                                outk, outv);
}